// CRF_20899310862347
// MI455X (gfx1250) — compile-verified
//
#include <hip/hip_runtime.h>

typedef __attribute__((ext_vector_type(2))) float v2f;
typedef __attribute__((ext_vector_type(8))) float v8f;

#define B_SZ   1024
#define T_SZ   512
#define K_SZ   64
#define ROWS   16     // batch rows per block
#define STRIDE 68     // LDS row stride in floats (bank-conflict-free padding)

// Raw-hardware transcendentals: inputs are guaranteed normal/positive
// (S >= ~0.5, sumP >= 1, exp args <= 0), so skip libm denorm guards.
__device__ __forceinline__ float fast_exp(float x) {
  return __builtin_amdgcn_exp2f(x * 1.4426950408889634f);   // v_exp_f32
}
__device__ __forceinline__ float fast_log(float x) {
  return __builtin_amdgcn_logf(x) * 0.6931471805599453f;    // v_log_f32
}

__global__ __launch_bounds__(128) void crf_forward_wmma(
    const float* __restrict__ feats,      // [B, T, K]
    const float* __restrict__ trans,      // [K, K]
    const int*   __restrict__ seq_lens,   // [B]
    float*       __restrict__ out)        // [B]
{
  __shared__ float lds_fv[ROWS * STRIDE];   // log-domain forward variables
  __shared__ float lds_P [ROWS * STRIDE];   // exp(fv - m), A-matrix source
  __shared__ float lds_m [ROWS];            // per-row max

  const int tid  = threadIdx.x;
  const int lane = tid & 31;
  const int wave = tid >> 5;
  const int r0   = blockIdx.x * ROWS;

  // ---------------------------------------------------------------
  // B-matrix registers (held for the whole kernel): wave w produces
  // output tags [n0, n0+16).  f32 4x16 B layout per chunk:
  // VGPR0 = {K0 | K2}, VGPR1 = {K1 | K3} across lane halves.
  // W[j][i] = exp(trans[i][j]).
  // ---------------------------------------------------------------
  const int n0   = wave * 16;
  const int half = lane >> 4;      // 0: lanes 0-15, 1: lanes 16-31
  const int lcol = lane & 15;
  const int icol = n0 + lcol;      // output tag index for this lane
  v2f Wreg[16];
  #pragma unroll
  for (int k0 = 0; k0 < 16; ++k0) {
    const int kb = k0 * 4 + half * 2;              // j (reduction) index
    Wreg[k0].x = fast_exp(trans[icol * K_SZ + kb]);
    Wreg[k0].y = fast_exp(trans[icol * K_SZ + kb + 1]);
  }

  // Phase-A mapping: 8 threads per batch row, 8 columns each.
  const int ar   = tid >> 3;            // row 0..15
  const int ac0  = (tid & 7) * 8;       // column base
  const int frow = (r0 + ar) * T_SZ * K_SZ;
  const int slen = seq_lens[r0 + ar];

  // D-row feats base for the WMMA phase (row = half*8 + v).
  const int dbase = (r0 + half * 8) * T_SZ * K_SZ + icol;

  // ---- init: fv_0 = feats[:, 0, :] ----
  #pragma unroll
  for (int i = 0; i < 8; ++i)
    lds_fv[ar * STRIDE + ac0 + i] = feats[frow + ac0 + i];
  __syncthreads();

  for (int t = 1; t <= T_SZ; ++t) {
    // ------------------------------------------------------------
    // Phase A on fv_{t-1}: row max, P = exp(fv-m), per-row sumP.
    // Output alpha written exactly when t-1 == seq_len-1.
    // ------------------------------------------------------------
    float vals[8];
    vals[0] = lds_fv[ar * STRIDE + ac0];
    float vmax = vals[0];
    #pragma unroll
    for (int i = 1; i < 8; ++i) {
      vals[i] = lds_fv[ar * STRIDE + ac0 + i];
      vmax = fmaxf(vmax, vals[i]);
    }
    #pragma unroll
    for (int m = 1; m <= 4; m <<= 1)
      vmax = fmaxf(vmax, __shfl_xor(vmax, m, 32));   // 8-lane row group

    float psum = 0.0f;
    #pragma unroll
    for (int i = 0; i < 8; ++i) {
      const float p = fast_exp(vals[i] - vmax);
      lds_P[ar * STRIDE + ac0 + i] = p;
      psum += p;
    }
    #pragma unroll
    for (int m = 1; m <= 4; m <<= 1)
      psum += __shfl_xor(psum, m, 32);

    if ((tid & 7) == 0) {
      lds_m[ar] = vmax;
      if (slen - 1 == t - 1)                          // lse_i(fv_{t-1})
        out[r0 + ar] = vmax + fast_log(psum);
    }
    if (t == T_SZ) break;                             // uniform exit

    // Prefetch feats 8 steps ahead of the dependent chain.
    if (t + 8 < T_SZ)
      __builtin_prefetch(&feats[frow + (t + 8) * K_SZ + ac0], 0, 1);

    __syncthreads();

    // ------------------------------------------------------------
    // Preload ALL A-matrix chunks (8 x ds_load_2addr_b64, one wait),
    // plus this step's feats and per-row maxima, BEFORE the WMMA
    // chain so their latency hides under the 16 dependent WMMAs.
    // ------------------------------------------------------------
    v2f Areg[16];
    #pragma unroll
    for (int k0 = 0; k0 < 16; ++k0)
      Areg[k0] = *(const v2f*)&lds_P[lcol * STRIDE + k0 * 4 + half * 2];

    float fnew[8];
    #pragma unroll
    for (int v = 0; v < 8; ++v)
      fnew[v] = feats[dbase + v * (T_SZ * K_SZ) + t * K_SZ];

    const float4 mlo = *(const float4*)&lds_m[half * 8];
    const float4 mhi = *(const float4*)&lds_m[half * 8 + 4];
    const float mr[8] = {mlo.x, mlo.y, mlo.z, mlo.w, mhi.x, mhi.y, mhi.z, mhi.w};

    // S = P(16x64) x Wtile(64x16): 16 back-to-back f32 16x16x4 WMMAs.
    v8f acc = {};
    #pragma unroll
    for (int k0 = 0; k0 < 16; ++k0)
      acc = __builtin_amdgcn_wmma_f32_16x16x4_f32(
          /*neg_a=*/false, Areg[k0], /*neg_b=*/false, Wreg[k0],
          /*c_mod=*/(short)0, acc, /*reuse_a=*/false, /*reuse_b=*/false);

    // fv_t = feat[t] + m + log(S); D layout: VGPR v -> row half*8 + v.
    #pragma unroll
    for (int v = 0; v < 8; ++v) {
      const int row = half * 8 + v;
      lds_fv[row * STRIDE + icol] = fnew[v] + mr[v] + fast_log(acc[v]);
    }
    __syncthreads();
  }
}

extern "C" void kernel_launch(void* const* d_in, const int* in_sizes, int n_in,
                              void* d_out, int out_size, void* d_ws, size_t ws_size,
                              hipStream_t stream) {
  const float* feats    = (const float*)d_in[0];
  const float* trans    = (const float*)d_in[1];
  const int*   seq_lens = (const int*)d_in[2];
  float*       out      = (float*)d_out;

  crf_forward_wmma<<<B_SZ / ROWS, 128, 0, stream>>>(feats, trans, seq_lens, out);
}